// Attention_6150393168186
// MI455X (gfx1250) — compile-verified
//
#include <hip/hip_runtime.h>
#include <math.h>

typedef __attribute__((ext_vector_type(2))) float v2f;
typedef __attribute__((ext_vector_type(8))) float v8f;

#define B_ 64
#define S_ 2048
#define R_ 512
#define KC 64
#define LSTRIDE 68   // 64 + 4 floats pad -> conflict-free WMMA A-fragment reads

// ---------------------------------------------------------------------------
// Pass 1: one streaming read of P. Per wave of 16 s-rows:
//   scores[row] = dot(P[row,:], W1[0:512])   (WMMA D column 0)
//   p2[row]     = dot(P[row,:], W2[0:512])   (WMMA D column 1)
// A = 16 rows x 4 K of P (staged through LDS), B = weights in columns 0/1.
// Each wave owns a private LDS tile; same-wave DS ops are in-order (ISA),
// so no block barriers are needed inside the K loop — only an IR-level fence.
// ---------------------------------------------------------------------------
__global__ __launch_bounds__(256) void pass1_wmma(
    const float* __restrict__ P, const float* __restrict__ W1,
    const float* __restrict__ W2, float* __restrict__ scores,
    float* __restrict__ p2)
{
  __shared__ float wlds[3 * R_];             // w1_p | w2_p | zeros (cols >= 2)
  __shared__ float tile[8 * 16 * LSTRIDE];   // per-wave 16 x 68 tiles

  const int tid  = threadIdx.x;
  const int wave = tid >> 5;
  const int lane = tid & 31;
  const int lr   = lane & 15;    // row (A) / column (B,D) index
  const int hi   = lane >> 4;    // K-half select

  // stage the two 512-float weight vectors + a zero region (one time)
  wlds[tid]        = W1[tid];
  wlds[tid + 256]  = W1[tid + 256];
  wlds[tid + 512]  = W2[tid];
  wlds[tid + 768]  = W2[tid + 256];
  wlds[tid + 1024] = 0.f;
  wlds[tid + 1280] = 0.f;

  const int rowBase = blockIdx.x * 128 + wave * 16;   // flat (b*S + s) row
  const float* prow = P + (size_t)rowBase * R_;
  float* tw = &tile[wave * 16 * LSTRIDE];

  // branchless B-fragment source: lanes >= 2 read the zero region
  const int wsel = (lr < 2) ? lr : 2;
  const float* wbase = &wlds[wsel * R_ + 2 * hi];
  // A-fragment source within this wave's tile
  const float* arow = &tw[lr * LSTRIDE + 2 * hi];

  v8f acc = {};
  __syncthreads();   // weights visible to all waves

  for (int k0 = 0; k0 < R_; k0 += KC) {
    // coalesced stage: 2 rows x 64 floats per iteration (b128 per lane)
    #pragma unroll
    for (int rr = 0; rr < 8; ++rr) {
      const int row = rr * 2 + hi;
      const int col = lr * 4;
      const float4 v = *(const float4*)(prow + (size_t)row * R_ + k0 + col);
      *(float4*)(&tw[row * LSTRIDE + col]) = v;
    }
    // wave-local: DS in-order guarantees the stores land before reads below
    __asm__ volatile("" ::: "memory");

    #pragma unroll
    for (int kk = 0; kk < KC; kk += 4) {
      // A fragment (16x4 f32): lanes 0-15 hold K=0,1; lanes 16-31 hold K=2,3
      v2f a;
      a.x = arow[kk];
      a.y = arow[kk + 1];
      // B fragment (4x16 f32): col 0 = w1_p, col 1 = w2_p, cols 2..15 = 0
      v2f b;
      b.x = wbase[k0 + kk];
      b.y = wbase[k0 + kk + 1];
      acc = __builtin_amdgcn_wmma_f32_16x16x4_f32(
          false, a, false, b, (short)0, acc, false, false);
    }
    __asm__ volatile("" ::: "memory");
  }

  // D: lane n (=lr, hi selects row half) holds column n; VGPR i = row i + 8*hi
  if (lr < 2) {
    float* dst = (lr == 0) ? scores : p2;
    #pragma unroll
    for (int i = 0; i < 8; ++i)
      dst[rowBase + hi * 8 + i] = acc[i];
  }
}

// ---------------------------------------------------------------------------
// Softmax over s per batch row; overwrites scores with normalized weights a.
// (The h.w1_h per-b constant cancels under softmax, so it is never computed.)
// ---------------------------------------------------------------------------
__global__ __launch_bounds__(256) void pass_softmax(float* __restrict__ scores)
{
  __shared__ float red[256];
  const int b = blockIdx.x;
  const int t = threadIdx.x;
  float* row = scores + (size_t)b * S_;

  float v[8];
  float m = -INFINITY;
  #pragma unroll
  for (int j = 0; j < 8; ++j) { v[j] = row[t + j * 256]; m = fmaxf(m, v[j]); }
  red[t] = m; __syncthreads();
  for (int off = 128; off > 0; off >>= 1) {
    if (t < off) red[t] = fmaxf(red[t], red[t + off]);
    __syncthreads();
  }
  m = red[0]; __syncthreads();

  float s = 0.f;
  #pragma unroll
  for (int j = 0; j < 8; ++j) { v[j] = __expf(v[j] - m); s += v[j]; }
  red[t] = s; __syncthreads();
  for (int off = 128; off > 0; off >>= 1) {
    if (t < off) red[t] += red[t + off];
    __syncthreads();
  }
  const float rs = 1.0f / red[0];
  #pragma unroll
  for (int j = 0; j < 8; ++j) row[t + j * 256] = v[j] * rs;
}

// ---------------------------------------------------------------------------
// Pass 2: second streaming read of P. Each block = one (b, 128-row s-chunk);
// thread t owns r-columns {2t, 2t+1}. Writes per-chunk partial c_t.
// ---------------------------------------------------------------------------
#define SCH 128
__global__ __launch_bounds__(256) void pass2_ct(
    const float* __restrict__ P, const float* __restrict__ a,
    float* __restrict__ part)
{
  __shared__ float aa[SCH];
  const int blk   = blockIdx.x;        // b*16 + chunk
  const int b     = blk >> 4;
  const int chunk = blk & 15;
  const int t     = threadIdx.x;
  const int s0    = chunk * SCH;

  if (t < SCH) aa[t] = a[(size_t)b * S_ + s0 + t];
  __syncthreads();

  const float* pp = P + ((size_t)b * S_ + s0) * R_ + 2 * t;
  float2 acc = make_float2(0.f, 0.f);
  #pragma unroll 4
  for (int s = 0; s < SCH; ++s) {
    const float2 p = *(const float2*)(pp + (size_t)s * R_);
    const float w = aa[s];
    acc.x += p.x * w;
    acc.y += p.y * w;
  }
  float* dst = part + (size_t)blk * R_ + 2 * t;
  dst[0] = acc.x;
  dst[1] = acc.y;
}

// ct[b,r] = sum over the 16 s-chunk partials (deterministic, no atomics)
__global__ __launch_bounds__(256) void reduce_ct(
    const float* __restrict__ part, float* __restrict__ ct)
{
  const int i = blockIdx.x * 256 + threadIdx.x;  // < B_*R_
  const int b = i >> 9;
  const int r = i & (R_ - 1);
  float s = 0.f;
  #pragma unroll
  for (int j = 0; j < 16; ++j) s += part[(size_t)(b * 16 + j) * R_ + r];
  ct[i] = s;
}

// cdot[b] = dot(c_t[b,:], w2_c)
__global__ __launch_bounds__(256) void pass_cdot(
    const float* __restrict__ ct, const float* __restrict__ W2,
    float* __restrict__ cdot)
{
  __shared__ float red[256];
  const int b = blockIdx.x;
  const int t = threadIdx.x;
  float s = ct[b * R_ + t] * W2[R_ + t] +
            ct[b * R_ + 256 + t] * W2[R_ + 256 + t];
  red[t] = s; __syncthreads();
  for (int off = 128; off > 0; off >>= 1) {
    if (t < off) red[t] += red[t + off];
    __syncthreads();
  }
  if (t == 0) cdot[b] = red[0];
}

// out[b,s] = p2[b,s] + cdot[b]
__global__ __launch_bounds__(256) void pass_out(
    const float* __restrict__ p2, const float* __restrict__ cdot,
    float* __restrict__ out)
{
  const int i = blockIdx.x * 256 + threadIdx.x;  // < B_*S_
  out[i] = p2[i] + cdot[i >> 11];
}

extern "C" void kernel_launch(void* const* d_in, const int* in_sizes, int n_in,
                              void* d_out, int out_size, void* d_ws, size_t ws_size,
                              hipStream_t stream)
{
  const float* P  = (const float*)d_in[0];
  // d_in[1] = h_t : unused (per-b constant cancels in softmax; w1_h too)
  const float* W1 = (const float*)d_in[2];
  const float* W2 = (const float*)d_in[3];
  float* out = (float*)d_out;

  float* ws     = (float*)d_ws;
  float* scores = ws;                     // B*S   (becomes softmax weights a)
  float* p2     = scores + B_ * S_;       // B*S
  float* part   = p2 + B_ * S_;           // B*16*R
  float* ct     = part + B_ * 16 * R_;    // B*R
  float* cdot   = ct + B_ * R_;           // B

  pass1_wmma  <<<(B_ * S_) / 128, 256, 0, stream>>>(P, W1, W2, scores, p2);
  pass_softmax<<<B_,              256, 0, stream>>>(scores);
  pass2_ct    <<<B_ * 16,         256, 0, stream>>>(P, scores, part);
  reduce_ct   <<<(B_ * R_) / 256, 256, 0, stream>>>(part, ct);
  pass_cdot   <<<B_,              256, 0, stream>>>(ct, W2, cdot);
  pass_out    <<<(B_ * S_) / 256, 256, 0, stream>>>(p2, cdot, out);
}